// GCNModel_23244363006342
// MI455X (gfx1250) — compile-verified
//
#include <hip/hip_runtime.h>
#include <hip/hip_bf16.h>

#define N_FEAT 256
#define HIDDEN 16
#define NCLS   2

typedef float v2f __attribute__((ext_vector_type(2)));
typedef float v8f __attribute__((ext_vector_type(8)));

__device__ __forceinline__ void atomic_add_f32(float* p, float v) {
    // lowers to global_atomic_add_f32 (no CAS loop)
    unsafeAtomicAdd(p, v);
}

// ---- degree / norm ------------------------------------------------------
__global__ void k_deg_init(float* __restrict__ deg, int n) {
    int i = blockIdx.x * blockDim.x + threadIdx.x;
    if (i < n) deg[i] = 1.0f;                 // self-loop contribution
}

__global__ void k_deg_edges(const int* __restrict__ dst, float* __restrict__ deg, int E) {
    int e = blockIdx.x * blockDim.x + threadIdx.x;
    if (e < E) atomic_add_f32(&deg[dst[e]], 1.0f);
}

__global__ void k_dinv(float* __restrict__ deg, int n) {
    int i = blockIdx.x * blockDim.x + threadIdx.x;
    if (i < n) deg[i] = rsqrtf(deg[i]);       // in place: deg -> dinv
}

// ---- layer 1 GEMM via V_WMMA_F32_16X16X4_F32 ---------------------------
// hw1 = x @ W1 ; agg1 = hw1 * dinv^2   (self-loop init fused into store)
__global__ void k_gemm1(const float* __restrict__ x, const float* __restrict__ W1,
                        const float* __restrict__ dinv,
                        float* __restrict__ hw1, float* __restrict__ agg1, int ntiles) {
    int lane = threadIdx.x & 31;
    int wave = threadIdx.x >> 5;
    int tile = blockIdx.x * (blockDim.x >> 5) + wave;
    if (tile >= ntiles) return;               // uniform per wave -> EXEC all-ones inside
    int half = lane >> 4;                     // 0: K pair {0,1}, 1: K pair {2,3}
    int l    = lane & 15;
    int row0 = tile * 16;

    // A 16x4 f32 layout (ISA 7.12.2): lanes 0-15 rows M=0..15 K={0,1}; lanes 16-31 K={2,3}
    const float* aptr = x  + (size_t)(row0 + l) * N_FEAT + 2 * half;
    // B 4x16 f32 (mirror): v0 = row k0+2*half, v1 = row k0+2*half+1, col = l
    const float* bptr = W1 + (size_t)(2 * half) * HIDDEN + l;

    v8f c = {};
    #pragma unroll 8
    for (int kk = 0; kk < N_FEAT / 4; ++kk) {
        v2f a = *(const v2f*)(aptr + kk * 4);     // 8B aligned
        v2f b;
        b.x = bptr[kk * 4 * HIDDEN];
        b.y = bptr[kk * 4 * HIDDEN + HIDDEN];
        c = __builtin_amdgcn_wmma_f32_16x16x4_f32(false, a, false, b,
                                                  (short)0, c, false, false);
    }

    // C/D layout: VGPR v, lanes 0-15 -> M=v, lanes 16-31 -> M=v+8; col = l
    int rbase = row0 + half * 8;
    #pragma unroll
    for (int v = 0; v < 8; ++v) {
        int r = rbase + v;
        float dval = c[v];
        float di   = dinv[r];
        size_t idx = (size_t)r * HIDDEN + l;
        hw1[idx]  = dval;
        agg1[idx] = dval * di * di;           // self-loop: norm = dinv[i]^2
    }
}

// scalar remainder tiles (unused for N=100000 but keeps the kernel general)
__global__ void k_gemm1_rem(const float* __restrict__ x, const float* __restrict__ W1,
                            const float* __restrict__ dinv,
                            float* __restrict__ hw1, float* __restrict__ agg1,
                            int row_lo, int n) {
    int t = blockIdx.x * blockDim.x + threadIdx.x;
    int r = row_lo + t / HIDDEN;
    int c = t % HIDDEN;
    if (r >= n) return;
    float acc = 0.0f;
    for (int k = 0; k < N_FEAT; ++k) acc += x[(size_t)r * N_FEAT + k] * W1[(size_t)k * HIDDEN + c];
    float di = dinv[r];
    size_t idx = (size_t)r * HIDDEN + c;
    hw1[idx]  = acc;
    agg1[idx] = acc * di * di;
}

// ---- layer 1 edge scatter: 16 lanes per edge ----------------------------
__global__ void k_edges1(const int* __restrict__ src, const int* __restrict__ dst,
                         const float* __restrict__ hw1, const float* __restrict__ dinv,
                         float* __restrict__ agg1, int E) {
    int t = blockIdx.x * blockDim.x + threadIdx.x;
    int e = t >> 4;
    int c = t & 15;
    if (e >= E) return;
    int s = src[e], d = dst[e];
    float norm = dinv[s] * dinv[d];
    atomic_add_f32(&agg1[(size_t)d * HIDDEN + c], hw1[(size_t)s * HIDDEN + c] * norm);
}

// ---- bias + ReLU + GEMM2 (16->2) + layer-2 self-loop init --------------
__global__ void k_layer2(const float* __restrict__ agg1, const float* __restrict__ b1,
                         const float* __restrict__ W2, const float* __restrict__ dinv,
                         float* __restrict__ hw2, float* __restrict__ agg2, int n) {
    int i = blockIdx.x * blockDim.x + threadIdx.x;
    if (i >= n) return;
    float h0 = 0.0f, h1 = 0.0f;
    #pragma unroll
    for (int c = 0; c < HIDDEN; ++c) {
        float v = agg1[(size_t)i * HIDDEN + c] + b1[c];
        v = fmaxf(v, 0.0f);
        h0 += v * W2[c * NCLS + 0];
        h1 += v * W2[c * NCLS + 1];
    }
    float di  = dinv[i];
    float di2 = di * di;
    hw2[(size_t)i * NCLS + 0]  = h0;
    hw2[(size_t)i * NCLS + 1]  = h1;
    agg2[(size_t)i * NCLS + 0] = h0 * di2;    // self-loop contribution
    agg2[(size_t)i * NCLS + 1] = h1 * di2;
}

// ---- layer 2 edge scatter: 1 thread per edge, 2 channels ----------------
__global__ void k_edges2(const int* __restrict__ src, const int* __restrict__ dst,
                         const float* __restrict__ hw2, const float* __restrict__ dinv,
                         float* __restrict__ agg2, int E) {
    int e = blockIdx.x * blockDim.x + threadIdx.x;
    if (e >= E) return;
    int s = src[e], d = dst[e];
    float norm = dinv[s] * dinv[d];
    atomic_add_f32(&agg2[(size_t)d * NCLS + 0], hw2[(size_t)s * NCLS + 0] * norm);
    atomic_add_f32(&agg2[(size_t)d * NCLS + 1], hw2[(size_t)s * NCLS + 1] * norm);
}

// ---- bias + log_softmax -------------------------------------------------
__global__ void k_final(const float* __restrict__ agg2, const float* __restrict__ b2,
                        float* __restrict__ out, int n) {
    int i = blockIdx.x * blockDim.x + threadIdx.x;
    if (i >= n) return;
    float l0 = agg2[(size_t)i * NCLS + 0] + b2[0];
    float l1 = agg2[(size_t)i * NCLS + 1] + b2[1];
    float m   = fmaxf(l0, l1);
    float lse = m + logf(expf(l0 - m) + expf(l1 - m));
    out[(size_t)i * NCLS + 0] = l0 - lse;
    out[(size_t)i * NCLS + 1] = l1 - lse;
}

extern "C" void kernel_launch(void* const* d_in, const int* in_sizes, int n_in,
                              void* d_out, int out_size, void* d_ws, size_t ws_size,
                              hipStream_t stream) {
    const float* x  = (const float*)d_in[0];
    const int*   ei = (const int*)d_in[1];   // [2, E]: row 0 = src, row 1 = dst
    const float* W1 = (const float*)d_in[2];
    const float* b1 = (const float*)d_in[3];
    const float* W2 = (const float*)d_in[4];
    const float* b2 = (const float*)d_in[5];
    float* out = (float*)d_out;

    int N = in_sizes[0] / N_FEAT;
    int E = in_sizes[1] / 2;
    const int* src = ei;
    const int* dst = ei + E;

    float* ws   = (float*)d_ws;
    float* dinv = ws;                               // N
    float* hw1  = dinv + (size_t)N;                 // 16N
    float* agg1 = hw1  + (size_t)N * HIDDEN;        // 16N
    float* hw2  = agg1 + (size_t)N * HIDDEN;        // 2N
    float* agg2 = hw2  + (size_t)N * NCLS;          // 2N

    const int tb = 256;
    k_deg_init <<<(N + tb - 1) / tb, tb, 0, stream>>>(dinv, N);
    k_deg_edges<<<(E + tb - 1) / tb, tb, 0, stream>>>(dst, dinv, E);
    k_dinv     <<<(N + tb - 1) / tb, tb, 0, stream>>>(dinv, N);

    int ntiles = N / 16;
    int wpb    = tb / 32;                            // waves (tiles) per block
    if (ntiles > 0)
        k_gemm1<<<(ntiles + wpb - 1) / wpb, tb, 0, stream>>>(x, W1, dinv, hw1, agg1, ntiles);
    int rem = N - ntiles * 16;
    if (rem > 0)
        k_gemm1_rem<<<(rem * HIDDEN + tb - 1) / tb, tb, 0, stream>>>(x, W1, dinv, hw1, agg1,
                                                                     ntiles * 16, N);

    long long t1 = (long long)E * HIDDEN;
    k_edges1<<<(int)((t1 + tb - 1) / tb), tb, 0, stream>>>(src, dst, hw1, dinv, agg1, E);
    k_layer2<<<(N + tb - 1) / tb, tb, 0, stream>>>(agg1, b1, W2, dinv, hw2, agg2, N);
    k_edges2<<<(E + tb - 1) / tb, tb, 0, stream>>>(src, dst, hw2, dinv, agg2, E);
    k_final <<<(N + tb - 1) / tb, tb, 0, stream>>>(agg2, b2, out, N);
}